// Constraint_Correlation_Layer_36155034698121
// MI455X (gfx1250) — compile-verified
//
#include <hip/hip_runtime.h>

typedef __attribute__((ext_vector_type(2))) float v2f;
typedef __attribute__((ext_vector_type(8))) float v8f;

#define N_      8
#define Z_      128
#define H_      160
#define W_      160
#define MD      9        // max_disp
#define RAD     4        // (MD+1)/2 - 1
#define ZC      16       // z-chunk through LDS
#define APITCH  168      // A row pitch (pad: avoids 2-way bank conflict)
#define BPITCH  176      // B row pitch: 160 + 2*RAD halo, padded to 176
#define NTHREADS 320     // 10 wave32 -> 10 x-tiles of 16 pixels
#define W4      (W_ / 4) // 40 b128 loads per 160-float row

// LDS floats: two (A|B) chunk buffers (double-buffered) + output staging [81][W]
#define A_FLOATS     (ZC * APITCH)
#define B_FLOATS     (MD * ZC * BPITCH)
#define CHUNK_FLOATS (A_FLOATS + B_FLOATS)
#define ST_FLOATS    (MD * MD * W_)
#define SMEM_BYTES   ((2 * CHUNK_FLOATS + ST_FLOATS) * sizeof(float))

// ---- CDNA5 async global->LDS (ASYNCcnt-tracked, no VGPR round trip) ----
__device__ __forceinline__ void async_b128(const float* lds_ptr, const float* gptr) {
  // low 32 bits of a flat shared-aperture address == wave-relative LDS byte addr
  unsigned lds_off = (unsigned)(size_t)lds_ptr;
  asm volatile("global_load_async_to_lds_b128 %0, %1, off"
               :: "v"(lds_off), "v"(gptr)
               : "memory");
}
__device__ __forceinline__ void wait_async0() {
  asm volatile("s_wait_asynccnt 0x0" ::: "memory");
}

// Issue the async fill of one (A|B) chunk buffer. B rows with off-image h+dy
// are skipped (block-uniform predicate); those LDS regions stay zero from the
// one-time zero fill. All transfers are 16B-aligned B128.
__device__ __forceinline__ void issue_chunk(float* buf, int zc, int tid, int h,
                                            const float* Abase, const float* Bbase) {
  // A: 16 z-rows x 160 cols
  for (int idx = tid; idx < ZC * W4; idx += NTHREADS) {
    int zz = idx / W4;
    int c4 = idx - zz * W4;
    async_b128(buf + zz * APITCH + c4 * 4,
               Abase + (size_t)(zc + zz) * (H_ * W_) + c4 * 4);
  }
  // B: 9 halo rows x 16 z x 160 valid cols (into halo-offset RAD)
  for (int idx = tid; idx < MD * ZC * W4; idx += NTHREADS) {
    int j  = idx / (ZC * W4);
    int r  = idx - j * (ZC * W4);
    int zz = r / W4;
    int c4 = r - zz * W4;
    int row = h + j - RAD;
    if (row >= 0 && row < H_) {
      async_b128(buf + A_FLOATS + (j * ZC + zz) * BPITCH + RAD + c4 * 4,
                 Bbase + (size_t)(zc + zz) * (H_ * W_) + (size_t)row * W_ + c4 * 4);
    }
  }
}

__global__ __launch_bounds__(NTHREADS)
void corr_volume_wmma(const float* __restrict__ A,
                      const float* __restrict__ B,
                      float* __restrict__ out) {
  extern __shared__ float smem[];
  float* buf0 = smem;                      // [A: ZC][APITCH] | [B: MD][ZC][BPITCH]
  float* buf1 = smem + CHUNK_FLOATS;
  float* St   = smem + 2 * CHUNK_FLOATS;   // [81][W_]

  const int nb   = blockIdx.x / H_;
  const int h    = blockIdx.x % H_;
  const int tid  = threadIdx.x;
  const int lane = tid & 31;
  const int wave = tid >> 5;        // 0..9 -> x-tile
  const int w0   = wave * 16;
  const int half = lane >> 4;       // 0: K=0,1   1: K=2,3
  const int nq   = lane & 15;       // M (for A) / N (for B) index

  const float* Abase = A + (size_t)nb * Z_ * (H_ * W_) + (size_t)h * W_;
  const float* Bbase = B + (size_t)nb * Z_ * (H_ * W_);

  // One-time zero fill of both B halo buffers (async loads only overwrite
  // the chunk-invariant valid interior afterwards).
  for (int idx = tid; idx < B_FLOATS; idx += NTHREADS) {
    buf0[A_FLOATS + idx] = 0.0f;
    buf1[A_FLOATS + idx] = 0.0f;
  }
  __syncthreads();

  const v8f zacc = {0.f, 0.f, 0.f, 0.f, 0.f, 0.f, 0.f, 0.f};
  v8f c1[MD], c2[MD];
#pragma unroll
  for (int j = 0; j < MD; ++j) { c1[j] = zacc; c2[j] = zacc; }

  // Prime the pipeline: fill buffer 0 with chunk 0.
  issue_chunk(buf0, 0, tid, h, Abase, Bbase);
  wait_async0();
  __syncthreads();

  for (int zc = 0; zc < Z_; zc += ZC) {
    float* cur = ((zc / ZC) & 1) ? buf1 : buf0;
    float* nxt = ((zc / ZC) & 1) ? buf0 : buf1;
    // Overlap: stream chunk zc+ZC into the other buffer while computing.
    if (zc + ZC < Z_) issue_chunk(nxt, zc + ZC, tid, h, Abase, Bbase);

    const float* As = cur;
    const float* Bs = cur + A_FLOATS;
#pragma unroll
    for (int kk = 0; kk < ZC; kk += 4) {
      const int kr = kk + 2 * half;          // lane's K rows: kr, kr+1
      v2f a;
      a.x = As[kr * APITCH + w0 + nq];
      a.y = As[(kr + 1) * APITCH + w0 + nq];
#pragma unroll
      for (int j = 0; j < MD; ++j) {
        const float* bj = Bs + (j * ZC + kr) * BPITCH;
        v2f b1, b2;
        b1.x = bj[w0 + nq];               // B tile 1: actual cols w0-4 .. w0+11
        b1.y = bj[BPITCH + w0 + nq];
        b2.x = bj[w0 + 16 + nq];          // B tile 2: actual cols w0+12 .. w0+27
        b2.y = bj[BPITCH + w0 + 16 + nq];
        c1[j] = __builtin_amdgcn_wmma_f32_16x16x4_f32(
                    false, a, false, b1, (short)0, c1[j], false, false);
        c2[j] = __builtin_amdgcn_wmma_f32_16x16x4_f32(
                    false, a, false, b2, (short)0, c2[j], false, false);
      }
    }
    // Own async issues done, then workgroup-wide publish of 'nxt'.
    wait_async0();
    __syncthreads();
  }

  // ---- scatter C[p][q] -> staging[k][w]; xshift s = q - p - 4 (+16 for tile2) ----
  const int mbase = 8 * half;  // C/D layout: VGPR v holds M = v + 8*(lane>=16), N = lane&15
#pragma unroll
  for (int j = 0; j < MD; ++j) {
#pragma unroll
    for (int v = 0; v < 8; ++v) {
      const int m  = mbase + v;
      const int s1 = nq - m - RAD;
      if (s1 >= -RAD && s1 <= RAD)
        St[((s1 + RAD) * MD + j) * W_ + w0 + m] = c1[j][v];
      const int s2 = nq - m + 16 - RAD;
      if (s2 >= -RAD && s2 <= RAD)
        St[((s2 + RAD) * MD + j) * W_ + w0 + m] = c2[j][v];
    }
  }
  __syncthreads();
  // ---- coalesced store: out[n, k, h, 0..159] ----
  float* obase = out + (size_t)nb * (MD * MD) * (H_ * W_) + (size_t)h * W_;
  for (int idx = tid; idx < MD * MD * W_; idx += NTHREADS) {
    int k = idx / W_;
    int w = idx - k * W_;
    obase[(size_t)k * (H_ * W_) + w] = St[idx];
  }
}

extern "C" void kernel_launch(void* const* d_in, const int* in_sizes, int n_in,
                              void* d_out, int out_size, void* d_ws, size_t ws_size,
                              hipStream_t stream) {
  const float* A = (const float*)d_in[0];
  const float* B = (const float*)d_in[1];
  float* out = (float*)d_out;
  (void)in_sizes; (void)n_in; (void)out_size; (void)d_ws; (void)ws_size;
  dim3 grid(N_ * H_);
  dim3 block(NTHREADS);
  corr_volume_wmma<<<grid, block, SMEM_BYTES, stream>>>(A, B, out);
}